// GNNModel_1683627180254
// MI455X (gfx1250) — compile-verified
//
#include <hip/hip_runtime.h>
#include <math.h>

typedef __attribute__((ext_vector_type(2))) float v2f;
typedef __attribute__((ext_vector_type(8))) float v8f;

#define KD 128  // inner dim of every GEMM in this model (IN_CH == HID_CH == 128)

// ---------------------------------------------------------------- degree prep
__global__ void k_init_deg(float* __restrict__ deg, int n) {
  int i = blockIdx.x * blockDim.x + threadIdx.x;
  if (i < n) deg[i] = 1.0f;  // self-loop contributes 1 to every node's degree
}

__global__ void k_count_deg(const int* __restrict__ dst, float* __restrict__ deg, int e) {
  int i = blockIdx.x * blockDim.x + threadIdx.x;
  if (i < e) unsafeAtomicAdd(&deg[dst[i]], 1.0f);  // hw global_atomic_add_f32
}

__global__ void k_dinv(float* __restrict__ deg, int n) {
  int i = blockIdx.x * blockDim.x + threadIdx.x;
  if (i < n) deg[i] = rsqrtf(deg[i]);  // deg >= 1 always (self-loops)
}

// ---------------------------------------------------------------- WMMA GEMM
// C[M,FOUT] = act(A)[M,128] @ B[128,FOUT], row-major. One wave per 16x16 C tile.
// V_WMMA_F32_16X16X4_F32, 32 K-steps of 4, two accumulators to break the RAW chain.
// f32 16x4 A fragment (ISA 7.12.2): lanes 0-15 -> K={k0,k0+1}, lanes 16-31 -> K={k0+2,k0+3},
// row M = lane%16. B fragment mirrors with col N = lane%16.
// Out-of-range columns are CLAMPED (not zeroed): column n of D depends only on
// column n of B, and clamped columns are never stored, so no zero-fill needed.
template <int FOUT, bool RELU_A>
__global__ void k_gemm_wmma(const float* __restrict__ A, const float* __restrict__ B,
                            float* __restrict__ C, int nrows) {
  const int lane  = threadIdx.x & 31;
  const int nl    = lane & 15;
  const int m     = blockIdx.x * 16 + nl;
  const int n     = blockIdx.y * 16 + nl;
  const int khalf = (lane >> 4) << 1;  // 0 for lanes 0-15, 2 for lanes 16-31
  const int mc    = (m < nrows) ? m : (nrows - 1);
  const int nc    = (n < FOUT) ? n : (FOUT - 1);

  const float* __restrict__ arow = A + (size_t)mc * KD + khalf;
  const float* __restrict__ bcol = B + (size_t)khalf * FOUT + nc;

  v8f acc0 = {}, acc1 = {};
#pragma unroll
  for (int k0 = 0; k0 < KD; k0 += 8) {
    v2f a0 = *(const v2f*)(arow + k0);      // A[m][k0+khalf .. +1], 8B aligned
    v2f a1 = *(const v2f*)(arow + k0 + 4);
    if (RELU_A) {
      a0.x = fmaxf(a0.x, 0.0f); a0.y = fmaxf(a0.y, 0.0f);
      a1.x = fmaxf(a1.x, 0.0f); a1.y = fmaxf(a1.y, 0.0f);
    }
    v2f b0, b1;
    b0.x = bcol[(k0 + 0) * FOUT];           // compile-time immediate offsets
    b0.y = bcol[(k0 + 1) * FOUT];
    b1.x = bcol[(k0 + 4) * FOUT];
    b1.y = bcol[(k0 + 5) * FOUT];
    acc0 = __builtin_amdgcn_wmma_f32_16x16x4_f32(false, a0, false, b0, (short)0, acc0,
                                                 false, false);
    acc1 = __builtin_amdgcn_wmma_f32_16x16x4_f32(false, a1, false, b1, (short)0, acc1,
                                                 false, false);
  }
  v8f acc = acc0 + acc1;

  if (n < FOUT) {
    const int rbase = blockIdx.x * 16 + ((lane >> 4) << 3);  // lanes 16-31: rows +8
#pragma unroll
    for (int v = 0; v < 8; ++v) {
      const int r = rbase + v;
      if (r < nrows) C[(size_t)r * FOUT + n] = acc[v];
    }
  }
}

// ------------------------------------------------- self-loop + bias (agg init)
// H[i,c] = T[i,c] * dinv[i]^2 + b[c]   (fully initializes H before edge atomics)
__global__ void k_selfloop_bias(const float4* __restrict__ T4, const float* __restrict__ dinv,
                                const float4* __restrict__ bias4, float4* __restrict__ H4,
                                int n, int f4) {
  int i = blockIdx.x * blockDim.x + threadIdx.x;
  if (i >= n * f4) return;
  int row = i / f4;
  int col = i - row * f4;
  float di = dinv[row];
  float w = di * di;
  float4 t = T4[i];
  float4 b = bias4[col];
  float4 h;
  h.x = t.x * w + b.x;
  h.y = t.y * w + b.y;
  h.z = t.z * w + b.z;
  h.w = t.w * w + b.w;
  H4[i] = h;
}

// ---------------------------------------------------------------- edge scatter
// One wave per edge; lane handles 4 consecutive channels (float4 gather + 4 f32 atomics).
__global__ void k_scatter(const float* __restrict__ T, const int* __restrict__ src,
                          const int* __restrict__ dstv, const float* __restrict__ dinv,
                          float* __restrict__ H, int e, int f) {
  int gid  = blockIdx.x * blockDim.x + threadIdx.x;
  int wid  = gid >> 5;
  int lane = gid & 31;
  if (wid >= e) return;
  int s = src[wid];
  int d = dstv[wid];
  float w = dinv[s] * dinv[d];
  int c = lane << 2;
  if (c < f) {  // f is a multiple of 4 (128 or 40)
    const float4 t = *(const float4*)(T + (size_t)s * f + c);
    float* hp = H + (size_t)d * f + c;
    unsafeAtomicAdd(hp + 0, t.x * w);
    unsafeAtomicAdd(hp + 1, t.y * w);
    unsafeAtomicAdd(hp + 2, t.z * w);
    unsafeAtomicAdd(hp + 3, t.w * w);
  }
}

// ---------------------------------------------------------------- log_softmax
// One wave per row; f <= 64 (here f = 40). In-place.
__global__ void k_logsoftmax(float* __restrict__ X, int n, int f) {
  int gid  = blockIdx.x * blockDim.x + threadIdx.x;
  int row  = gid >> 5;
  int lane = gid & 31;
  if (row >= n) return;
  float* xr = X + (size_t)row * f;
  float v0 = (lane < f) ? xr[lane] : -INFINITY;
  float v1 = (lane + 32 < f) ? xr[lane + 32] : -INFINITY;
  float m = fmaxf(v0, v1);
#pragma unroll
  for (int off = 16; off > 0; off >>= 1) m = fmaxf(m, __shfl_xor(m, off, 32));
  float s = ((lane < f) ? expf(v0 - m) : 0.0f) + ((lane + 32 < f) ? expf(v1 - m) : 0.0f);
#pragma unroll
  for (int off = 16; off > 0; off >>= 1) s += __shfl_xor(s, off, 32);
  float lse = m + logf(s);
  if (lane < f) xr[lane] = v0 - lse;
  if (lane + 32 < f) xr[lane + 32] = v1 - lse;
}

// ================================================================ launch
extern "C" void kernel_launch(void* const* d_in, const int* in_sizes, int n_in,
                              void* d_out, int out_size, void* d_ws, size_t ws_size,
                              hipStream_t stream) {
  const float* x  = (const float*)d_in[0];
  const int*   ei = (const int*)d_in[1];
  const float* W1 = (const float*)d_in[2];
  const float* b1 = (const float*)d_in[3];
  const float* W2 = (const float*)d_in[4];
  const float* b2 = (const float*)d_in[5];
  const float* W3 = (const float*)d_in[6];
  const float* b3 = (const float*)d_in[7];

  const int HID = in_sizes[3];           // 128
  const int OUT = in_sizes[7];           // 40
  const int N   = in_sizes[0] / KD;      // 50000
  const int E   = in_sizes[1] / 2;       // 800000
  const int* src = ei;
  const int* dst = ei + E;

  float* ws   = (float*)d_ws;
  float* dinv = ws;
  size_t o1 = ((size_t)N + 63) & ~(size_t)63;
  float* T = ws + o1;
  size_t o2 = ((size_t)N * HID + 63) & ~(size_t)63;
  float* H = T + o2;
  float* out = (float*)d_out;

  const int TB = 256;
  // ---- normalization coefficients
  k_init_deg<<<(N + TB - 1) / TB, TB, 0, stream>>>(dinv, N);
  k_count_deg<<<(E + TB - 1) / TB, TB, 0, stream>>>(dst, dinv, E);
  k_dinv<<<(N + TB - 1) / TB, TB, 0, stream>>>(dinv, N);

  const int gm = (N + 15) / 16;
  const int scat_blocks = (int)(((size_t)E * 32 + TB - 1) / TB);

  // ---- layer 1: T = x@W1 ; H = D^-1/2 A_hat D^-1/2 T + b1   (relu deferred)
  {
    dim3 grid(gm, (HID + 15) / 16);
    k_gemm_wmma<128, false><<<grid, 32, 0, stream>>>(x, W1, T, N);
    k_selfloop_bias<<<(N * (HID / 4) + TB - 1) / TB, TB, 0, stream>>>(
        (const float4*)T, dinv, (const float4*)b1, (float4*)H, N, HID / 4);
    k_scatter<<<scat_blocks, TB, 0, stream>>>(T, src, dst, dinv, H, E, HID);
  }
  // ---- layer 2: T = relu(H)@W2 ; H = agg(T) + b2   (relu fused into GEMM A-load)
  {
    dim3 grid(gm, (HID + 15) / 16);
    k_gemm_wmma<128, true><<<grid, 32, 0, stream>>>(H, W2, T, N);
    k_selfloop_bias<<<(N * (HID / 4) + TB - 1) / TB, TB, 0, stream>>>(
        (const float4*)T, dinv, (const float4*)b2, (float4*)H, N, HID / 4);
    k_scatter<<<scat_blocks, TB, 0, stream>>>(T, src, dst, dinv, H, E, HID);
  }
  // ---- layer 3: T = relu(H)@W3 (N x 40) ; out = agg(T) + b3 ; log_softmax
  {
    dim3 grid(gm, (OUT + 15) / 16);
    k_gemm_wmma<40, true><<<grid, 32, 0, stream>>>(H, W3, T, N);
    k_selfloop_bias<<<(N * (OUT / 4) + TB - 1) / TB, TB, 0, stream>>>(
        (const float4*)T, dinv, (const float4*)b3, (float4*)out, N, OUT / 4);
    k_scatter<<<scat_blocks, TB, 0, stream>>>(T, src, dst, dinv, out, E, OUT);
    k_logsoftmax<<<(int)(((size_t)N * 32 + TB - 1) / TB), TB, 0, stream>>>(out, N, OUT);
  }
}